// RelativeAttention_78271484002313
// MI455X (gfx1250) — compile-verified
//
#include <hip/hip_runtime.h>
#include <math.h>

// Problem dims (compile-time constants from the reference)
#define SB     4
#define SEQ    2048
#define DMIN   512
#define NH     8
#define DH     64
#define DOUTC  512           // NH * DH
#define NSINK  2
#define WWIN   8
#define SP     (NSINK + SEQ) // 2050 tokens incl. sink
#define PROWS  (NSINK + WWIN)// 10 softmax rows
#define MTOT   (SB * SP)     // 8200 GEMM rows

typedef float v2f __attribute__((ext_vector_type(2)));
typedef float v8f __attribute__((ext_vector_type(8)));

// GEMM tiling: 64x128 tile per 256-thread block (8 waves), K-chunks of 32,
// double-buffered LDS.
#define TM 64
#define TN 128
#define TK 32
#define ASTR (TK + 4)   // 36 floats = 144 B (16B multiple -> b128 LDS ops OK)
#define BSTR (TN + 4)   // 132 floats = 528 B (16B multiple)

// ---------------------------------------------------------------------------
// Phase 1: Q/K/V projection GEMM, f32 WMMA 16x16x4, double-buffered LDS.
//   blockIdx.x -> M tile, blockIdx.y -> N tile, blockIdx.z -> {0:Q, 1:K(+bk), 2:V}
//   Output layout per matrix: [MTOT, DOUTC] row-major in workspace.
// ---------------------------------------------------------------------------
__global__ __launch_bounds__(256)
void qkv_gemm_kernel(const float* __restrict__ x, const float* __restrict__ sink,
                     const float* __restrict__ Wq, const float* __restrict__ Wk,
                     const float* __restrict__ Wv, const float* __restrict__ bk,
                     float* __restrict__ ws) {
    __shared__ float As[2][TM * ASTR];
    __shared__ float Bs[2][TK * BSTR];

    const int tid  = threadIdx.x;
    const int lane = tid & 31;
    const int wave = tid >> 5;
    const int wm   = wave & 3;   // wave row band: 16 rows
    const int wn   = wave >> 2;  // wave col band: 64 cols
    const int lr   = lane & 15;  // row/col within fragment
    const int lh   = lane >> 4;  // half-wave -> K split per ISA layout

    const int m0 = blockIdx.x * TM;
    const int n0 = blockIdx.y * TN;
    const int z  = blockIdx.z;

    const float* Wm = (z == 0) ? Wq : (z == 1) ? Wk : Wv;
    float* out = ws + (size_t)z * MTOT * DOUTC;

    // ---- per-thread load/store geometry (fixed across the K loop) ----
    // A tile: 64 rows x 32 k = 512 float4, 2 per thread; row-level sink/x select.
    const float* aSrc[2];
    int aLds[2];
    #pragma unroll
    for (int i = 0; i < 2; ++i) {
        int e  = tid + i * 256;
        int r  = e >> 3;         // 0..63
        int c4 = e & 7;          // float4 column 0..7
        int m = m0 + r; if (m >= MTOT) m = MTOT - 1;   // clamp tail (EXEC all-1s)
        int b = m / SP;
        int t = m - b * SP;
        aSrc[i] = (t < NSINK)
            ? sink + ((size_t)b * NSINK + t) * DMIN + c4 * 4
            : x + ((size_t)b * SEQ + (t - NSINK)) * DMIN + c4 * 4;
        aLds[i] = r * ASTR + c4 * 4;
    }
    // B tile: 32 k x 128 n = 1024 float4, 4 per thread.
    const float* bSrc[4];
    int bLds[4];
    #pragma unroll
    for (int i = 0; i < 4; ++i) {
        int e  = tid + i * 256;
        int r  = e >> 5;         // 0..31
        int c4 = e & 31;         // float4 column 0..31
        bSrc[i] = Wm + (size_t)r * DOUTC + n0 + c4 * 4;
        bLds[i] = r * BSTR + c4 * 4;
    }

    float4 aReg[2], bReg[4];
    auto loadTile = [&](int k0) {   // global -> regs (b128, overlaps compute)
        #pragma unroll
        for (int i = 0; i < 2; ++i) aReg[i] = *(const float4*)(aSrc[i] + k0);
        #pragma unroll
        for (int i = 0; i < 4; ++i) bReg[i] = *(const float4*)(bSrc[i] + (size_t)k0 * DOUTC);
    };
    auto storeTile = [&](int buf) { // regs -> LDS (b128)
        #pragma unroll
        for (int i = 0; i < 2; ++i) *(float4*)&As[buf][aLds[i]] = aReg[i];
        #pragma unroll
        for (int i = 0; i < 4; ++i) *(float4*)&Bs[buf][bLds[i]] = bReg[i];
    };

    v8f acc[4];
    #pragma unroll
    for (int i = 0; i < 4; ++i)
        #pragma unroll
        for (int e = 0; e < 8; ++e) acc[i][e] = 0.0f;

    // ---- pipelined main loop ----
    loadTile(0);
    storeTile(0);
    __syncthreads();
    int cur = 0;

    for (int k0 = 0; k0 < DMIN; k0 += TK) {
        const int nk = k0 + TK;
        const bool more = (nk < DMIN);
        if (more) loadTile(nk);                       // issue next tile early
        if (k0 + 2 * TK < DMIN)                       // stream hint 2 chunks ahead
            __builtin_prefetch(bSrc[0] + (size_t)(k0 + 2 * TK) * DOUTC, 0, 1);

        #pragma unroll
        for (int kk = 0; kk < TK; kk += 4) {
            // A fragment 16x4 f32: lane(l) -> m = l&15, VGPR j -> k = j + 2*(l>>4)
            v2f a;
            a[0] = As[cur][(wm * 16 + lr) * ASTR + kk + 2 * lh + 0];
            a[1] = As[cur][(wm * 16 + lr) * ASTR + kk + 2 * lh + 1];
            #pragma unroll
            for (int nt = 0; nt < 4; ++nt) {
                // B fragment 4x16 f32: lane -> n = l&15, VGPR j -> k = j + 2*(l>>4)
                v2f bf;
                bf[0] = Bs[cur][(kk + 2 * lh + 0) * BSTR + wn * 64 + nt * 16 + lr];
                bf[1] = Bs[cur][(kk + 2 * lh + 1) * BSTR + wn * 64 + nt * 16 + lr];
                acc[nt] = __builtin_amdgcn_wmma_f32_16x16x4_f32(
                    /*neg_a=*/false, a, /*neg_b=*/false, bf,
                    /*c_mod=*/(short)0, acc[nt],
                    /*reuse_a=*/false, /*reuse_b=*/false);
            }
        }

        if (more) { storeTile(cur ^ 1); cur ^= 1; }
        __syncthreads();
    }

    // Store D: VGPR v holds (m = v + 8*(lane>>4), n = lane&15)
    #pragma unroll
    for (int nt = 0; nt < 4; ++nt) {
        int n = n0 + wn * 64 + nt * 16 + lr;
        float bias = (z == 1) ? bk[n] : 0.0f;
        #pragma unroll
        for (int v = 0; v < 8; ++v) {
            int m = m0 + wm * 16 + v + 8 * lh;
            if (m < MTOT)
                out[(size_t)m * DOUTC + n] = acc[nt][v] + bias;
        }
    }
}

// ---------------------------------------------------------------------------
// Phase 2: skewed relative attention. One thread per (b, s, c) output element.
//   10-way softmax over {NS sink rows, W window rows}; channels contiguous
//   so every global access is coalesced.
// ---------------------------------------------------------------------------
__global__ __launch_bounds__(256)
void relattn_kernel(const float* __restrict__ ws, const float* __restrict__ position,
                    float* __restrict__ out) {
    const float* Q = ws;
    const float* K = ws + (size_t)MTOT * DOUTC;
    const float* V = ws + 2 * (size_t)MTOT * DOUTC;

    int idx = blockIdx.x * blockDim.x + threadIdx.x;
    int c  = idx & (DOUTC - 1);
    int bs = idx >> 9;              // / DOUTC
    int s  = bs & (SEQ - 1);
    int b  = bs >> 11;              // / SEQ
    if (b >= SB) return;

    int d = c & (DH - 1);
    const float* pos = position + d * PROWS;   // position[1,1,DH,NS+W,1]

    const size_t base = (size_t)b * SP * DOUTC;
    const float kv = K[base + (size_t)(NSINK + s) * DOUTC + c];

    float sc[PROWS], vv[PROWS];
    #pragma unroll
    for (int p = 0; p < NSINK; ++p) {
        sc[p] = (Q[base + (size_t)p * DOUTC + c] + pos[p]) * kv;
        vv[p] = V[base + (size_t)p * DOUTC + c];
    }
    #pragma unroll
    for (int r = 0; r < WWIN; ++r) {
        int t = s + r;
        bool valid = (t < SEQ);
        int tc = valid ? t : (SEQ - 1);
        size_t row = base + (size_t)(NSINK + tc) * DOUTC + c;
        float qv = Q[row] + pos[NSINK + r];
        sc[NSINK + r] = valid ? (qv * kv) : -INFINITY;
        vv[NSINK + r] = valid ? V[row] : 0.0f;
    }

    float mx = sc[0];
    #pragma unroll
    for (int p = 1; p < PROWS; ++p) mx = fmaxf(mx, sc[p]);
    float den = 0.0f, num = 0.0f;
    #pragma unroll
    for (int p = 0; p < PROWS; ++p) {
        float e = __expf(sc[p] - mx);   // exp(-inf) == 0 kills masked rows
        den += e;
        num += e * vv[p];
    }
    out[(size_t)b * SEQ * DOUTC + (size_t)s * DOUTC + c] = num / den;
}

// ---------------------------------------------------------------------------
extern "C" void kernel_launch(void* const* d_in, const int* in_sizes, int n_in,
                              void* d_out, int out_size, void* d_ws, size_t ws_size,
                              hipStream_t stream) {
    // setup_inputs order: x, sink, Wk, bk, Wq, Wv, position (all float32)
    const float* x        = (const float*)d_in[0];
    const float* sink     = (const float*)d_in[1];
    const float* Wk       = (const float*)d_in[2];
    const float* bk       = (const float*)d_in[3];
    const float* Wq       = (const float*)d_in[4];
    const float* Wv       = (const float*)d_in[5];
    const float* position = (const float*)d_in[6];
    float* out = (float*)d_out;
    float* ws  = (float*)d_ws;   // needs 3 * MTOT * DOUTC * 4B ≈ 50.4 MB

    dim3 gA((MTOT + TM - 1) / TM, DOUTC / TN, 3);   // 129 x 4 x 3
    qkv_gemm_kernel<<<gA, 256, 0, stream>>>(x, sink, Wq, Wk, Wv, bk, ws);

    int total = SB * SEQ * DOUTC;                   // 4,194,304
    relattn_kernel<<<total / 256, 256, 0, stream>>>(ws, position, out);
}